// TripletLoss_57200374448277
// MI455X (gfx1250) — compile-verified
//
#include <hip/hip_runtime.h>
#include <hip/hip_bf16.h>
#include <math.h>

typedef __attribute__((ext_vector_type(2))) float v2f;
typedef __attribute__((ext_vector_type(8))) float v8f;

#define N 384
#define K 256
#define NT 24           // N/16 tiles
#define MARGIN 0.3f

// ---- workspace layout (float indices) ----
#define OFF_MAXD   0        // 1 uint (float bits of max_d)
#define OFF_SQ     64       // 384
#define OFF_HP     448      // 384
#define OFF_HN     832      // 384
#define OFF_HASP   1216     // 384
#define NPAIRBLK   18432    // (384*384)/8 waves-per-block
#define OFF_PARTS  1600     // 18432
#define OFF_PARTC  (1600 + NPAIRBLK)
#define OFF_D      (1600 + 2*NPAIRBLK)   // 384*384

// -------- 0: zero accumulators (ws is poisoned by harness) --------
__global__ void tl_init_kernel(float* W) {
  if (threadIdx.x == 0) {
    ((unsigned*)W)[OFF_MAXD] = 0u;   // float bits of 0.0f; valid for max of non-neg
  }
}

// -------- 1: squared norms sq[i] = dot(e_i, e_i) --------
__global__ __launch_bounds__(128) void tl_sq_kernel(const float* __restrict__ e,
                                                    float* __restrict__ sq) {
  int i = blockIdx.x * 128 + threadIdx.x;
  if (i < N) {
    const float* r = e + i * K;
    float s = 0.0f;
    for (int k = 0; k < K; ++k) s = fmaf(r[k], r[k], s);
    sq[i] = s;
  }
}

// -------- 2: WMMA Gram tiles -> pairwise Euclidean distances --------
// One wave per 16x16 tile of D. A = rows tileI, B = rows tileJ (B = A^T block).
// f32 WMMA 16x16x4: A 16x4 -> 2 VGPRs/lane; lanes 0-15 hold K=k..k+1,
// lanes 16-31 hold K=k+2..k+3 (ISA 7.12.2 32-bit A layout). B mirrors (4x16).
__global__ __launch_bounds__(32) void tl_dist_kernel(const float* __restrict__ e,
                                                     const float* __restrict__ sq,
                                                     float* __restrict__ D) {
  const int tileI = blockIdx.y;
  const int tileJ = blockIdx.x;
  const int lane  = threadIdx.x;
  const int half  = lane >> 4;       // 0: lanes 0-15, 1: lanes 16-31
  const int l16   = lane & 15;

  const float* arow = e + (tileI * 16 + l16) * K;  // A-matrix row (M = l16)
  const float* brow = e + (tileJ * 16 + l16) * K;  // B-matrix col (N = l16)

  v8f c = {};
  for (int k = 0; k < K; k += 4) {
    const int ka = k + half * 2;
    v2f a, b;
    a.x = arow[ka]; a.y = arow[ka + 1];
    b.x = brow[ka]; b.y = brow[ka + 1];
    // 8 args: (neg_a, A, neg_b, B, c_mod, C, reuse_a, reuse_b)
    c = __builtin_amdgcn_wmma_f32_16x16x4_f32(false, a, false, b, (short)0, c,
                                              false, false);
  }

  // C/D layout: VGPR r, lanes 0-15 -> M=r, N=lane; lanes 16-31 -> M=r+8, N=lane-16
  const int col = tileJ * 16 + l16;
  const float sqc = sq[col];
#pragma unroll
  for (int r = 0; r < 8; ++r) {
    const int row = tileI * 16 + r + half * 8;
    float s = sq[row] + sqc - 2.0f * c[r];
    s = fmaxf(s, 0.0f);
    const float dd = (s > 0.0f) ? sqrtf(s) : 0.0f;  // matches EPS/zmask semantics
    D[row * N + col] = dd;
  }
}

// -------- 3: global max of D (bit-pattern atomicMax; D >= 0) --------
__global__ __launch_bounds__(256) void tl_maxd_kernel(const float* __restrict__ D,
                                                      unsigned* __restrict__ maxbits) {
  __shared__ float sm[8];
  const int idx = blockIdx.x * 256 + threadIdx.x;
  float v = (idx < N * N) ? D[idx] : 0.0f;
#pragma unroll
  for (int off = 16; off > 0; off >>= 1) v = fmaxf(v, __shfl_xor(v, off, 32));
  if ((threadIdx.x & 31) == 0) sm[threadIdx.x >> 5] = v;
  __syncthreads();
  if (threadIdx.x == 0) {
    float m = sm[0];
#pragma unroll
    for (int w = 1; w < 8; ++w) m = fmaxf(m, sm[w]);
    atomicMax(maxbits, __float_as_uint(m));   // monotone for non-negative floats
  }
}

// -------- 4: per-row hardest positive / hardest negative (hard-loss fallback) --------
__global__ __launch_bounds__(32) void tl_hardrow_kernel(const float* __restrict__ D,
                                                        const int* __restrict__ labels,
                                                        const unsigned* __restrict__ maxbits,
                                                        float* __restrict__ hp,
                                                        float* __restrict__ hn,
                                                        float* __restrict__ hasPos) {
  const int i    = blockIdx.x;
  const int lane = threadIdx.x;
  const float maxd = __uint_as_float(maxbits[0]);
  const int li = labels[i];
  float hpv = 0.0f;          // (apf*d).max: zeros where mask off, so 0 is the identity
  float hnv = 3.4e38f;
  float cnt = 0.0f;
  const float* Di = D + i * N;
  for (int j = lane; j < N; j += 32) {
    const float dv = Di[j];
    const bool eq = (labels[j] == li);
    if (eq && (j != i)) { hpv = fmaxf(hpv, dv); cnt += 1.0f; }
    const float term = eq ? (dv + maxd) : dv;   // d + max_d*(1-anf)
    hnv = fminf(hnv, term);
  }
#pragma unroll
  for (int off = 16; off > 0; off >>= 1) {
    hpv = fmaxf(hpv, __shfl_xor(hpv, off, 32));
    hnv = fminf(hnv, __shfl_xor(hnv, off, 32));
    cnt += __shfl_xor(cnt, off, 32);
  }
  if (lane == 0) {
    hp[i] = hpv;
    hn[i] = hnv;
    hasPos[i] = (cnt > 0.0f) ? 1.0f : 0.0f;
  }
}

// -------- 5: semi-hard accumulation: one wave per (i,j) pair, lanes over k --------
__global__ __launch_bounds__(256) void tl_semi_kernel(const float* __restrict__ D,
                                                      const int* __restrict__ labels,
                                                      float* __restrict__ partS,
                                                      float* __restrict__ partC) {
  __shared__ float bs[8], bc[8];
  const int lane = threadIdx.x & 31;
  const int wid  = threadIdx.x >> 5;
  const int pair = blockIdx.x * 8 + wid;
  const int i = pair / N;
  const int j = pair - i * N;

  float s = 0.0f, c = 0.0f;
  const int li = labels[i];
  if ((labels[j] == li) && (i != j)) {       // ap_mask
    const float ap = D[i * N + j];
    const float* Di = D + i * N;
    for (int k = lane; k < N; k += 32) {
      if (labels[k] != li) {                 // an_mask
        const float an   = Di[k];
        const float loss = ap - an + MARGIN;
        if ((an > ap) && (loss > 0.0f)) { s += loss; c += 1.0f; }
      }
    }
  }
#pragma unroll
  for (int off = 16; off > 0; off >>= 1) {
    s += __shfl_xor(s, off, 32);
    c += __shfl_xor(c, off, 32);
  }
  if (lane == 0) { bs[wid] = s; bc[wid] = c; }
  __syncthreads();
  if (threadIdx.x == 0) {
    float ts = 0.0f, tc = 0.0f;
#pragma unroll
    for (int w = 0; w < 8; ++w) { ts += bs[w]; tc += bc[w]; }
    partS[blockIdx.x] = ts;
    partC[blockIdx.x] = tc;
  }
}

// -------- 6: final deterministic combine --------
__global__ __launch_bounds__(384) void tl_final_kernel(const float* __restrict__ hp,
                                                       const float* __restrict__ hn,
                                                       const float* __restrict__ hasPos,
                                                       const float* __restrict__ partS,
                                                       const float* __restrict__ partC,
                                                       float* __restrict__ out) {
  __shared__ float rS[12], rC[12], rTL[12], rV[12];
  const int lane = threadIdx.x & 31;
  const int wid  = threadIdx.x >> 5;

  float ls = 0.0f, lc = 0.0f;
  for (int p = threadIdx.x; p < NPAIRBLK; p += 384) { ls += partS[p]; lc += partC[p]; }

  const int i = threadIdx.x;               // 384 threads == 384 rows
  const float tl = fmaxf(hp[i] - hn[i] + MARGIN, 0.0f);
  float ltl = tl * hasPos[i];
  float lv  = hasPos[i];

#pragma unroll
  for (int off = 16; off > 0; off >>= 1) {
    ls  += __shfl_xor(ls, off, 32);
    lc  += __shfl_xor(lc, off, 32);
    ltl += __shfl_xor(ltl, off, 32);
    lv  += __shfl_xor(lv, off, 32);
  }
  if (lane == 0) { rS[wid] = ls; rC[wid] = lc; rTL[wid] = ltl; rV[wid] = lv; }
  __syncthreads();
  if (threadIdx.x == 0) {
    float tS = 0.0f, tC = 0.0f, tTL = 0.0f, tV = 0.0f;
#pragma unroll
    for (int w = 0; w < 12; ++w) { tS += rS[w]; tC += rC[w]; tTL += rTL[w]; tV += rV[w]; }
    const float hard = (tV > 0.0f) ? (tTL / fmaxf(tV, 1.0f)) : 0.0f;
    const float semi = tS / fmaxf(tC, 1.0f);
    out[0] = (tC > 0.0f) ? semi : hard;
  }
}

extern "C" void kernel_launch(void* const* d_in, const int* in_sizes, int n_in,
                              void* d_out, int out_size, void* d_ws, size_t ws_size,
                              hipStream_t stream) {
  const float* e      = (const float*)d_in[0];
  const int*   labels = (const int*)d_in[1];
  float* out = (float*)d_out;
  float* W   = (float*)d_ws;

  unsigned* maxbits = (unsigned*)W + OFF_MAXD;
  float* sq     = W + OFF_SQ;
  float* hp     = W + OFF_HP;
  float* hn     = W + OFF_HN;
  float* hasPos = W + OFF_HASP;
  float* partS  = W + OFF_PARTS;
  float* partC  = W + OFF_PARTC;
  float* D      = W + OFF_D;

  tl_init_kernel<<<1, 32, 0, stream>>>(W);
  tl_sq_kernel<<<3, 128, 0, stream>>>(e, sq);
  tl_dist_kernel<<<dim3(NT, NT), 32, 0, stream>>>(e, sq, D);
  tl_maxd_kernel<<<(N * N + 255) / 256, 256, 0, stream>>>(D, maxbits);
  tl_hardrow_kernel<<<N, 32, 0, stream>>>(D, labels, maxbits, hp, hn, hasPos);
  tl_semi_kernel<<<NPAIRBLK, 256, 0, stream>>>(D, labels, partS, partC);
  tl_final_kernel<<<1, 384, 0, stream>>>(hp, hn, hasPos, partS, partC, out);
}